// CustomParameterTransform_2491081031994
// MI455X (gfx1250) — compile-verified
//
#include <hip/hip_runtime.h>
#include <cmath>

// Problem constants (match reference: B=256, N_SRC=10, NMC=16, LL=128)
#define NBATCH 256
#define NSRC   10
#define NMC_   16
#define L_     128
#define PLANE   (L_ * L_)                  // 16384 floats (64 KB) per (b,c) plane
#define HALF    (PLANE / 2)                // 8192 floats (32 KB) per block tile
#define HALF_V4 (HALF / 4)                 // 2048 float4 per tile
#define NBLOCKS (NBATCH * 2 * NMC_ * 2)    // 16384 blocks (2 per plane)

typedef float v4f __attribute__((ext_vector_type(4)));
typedef int   v4i __attribute__((ext_vector_type(4)));

// gfx1250 async LDS->global store path (confirmed lowering in round 1),
// guarded so the file compiles on any toolchain.
#if defined(__gfx1250__) && \
    __has_builtin(__builtin_amdgcn_global_store_async_from_lds_b128) && \
    __has_builtin(__builtin_amdgcn_s_wait_asynccnt)
#define USE_ASYNC_LDS 1
#else
#define USE_ASYNC_LDS 0
#endif

// ---------------------------------------------------------------------------
// Fused fill + scatter. Each 256-thread block owns one 32 KB half of a
// (b, c) plane:
//   1) stage the base value (1.0 for c<16, 0.0 for c>=16) into LDS,
//   2) lanes 0..9 bin batch b's triplets and patch matching scatter points
//      (0.0 into the 1-z half, 1.0 into the z half) directly in the tile,
//   3) async-DMA the finished tile LDS -> HBM (ASYNCcnt-tracked b128 ops).
// ---------------------------------------------------------------------------
__global__ void __launch_bounds__(256) fused_kernel(const float* __restrict__ coord,
                                                    const float* __restrict__ lows,
                                                    const float* __restrict__ highs,
                                                    float* __restrict__ out) {
  const int half  = blockIdx.x & 1;
  const int plane = blockIdx.x >> 1;        // b*32 + c
  const int c     = plane & 31;
  const int b     = plane >> 5;
  const int mc    = c & (NMC_ - 1);
  const float v   = (c < NMC_) ? 1.0f : 0.0f;
  const float pv  = 1.0f - v;               // scatter patch value

  __shared__ v4f s_half[HALF_V4];           // 32 KB staging tile
  v4f v4 = {v, v, v, v};
#pragma unroll
  for (int i = 0; i < 8; ++i)
    s_half[i * 256 + threadIdx.x] = v4;
  __syncthreads();

  if (threadIdx.x < NSRC) {
    const float* tr = coord + (size_t)(b * 3 * NSRC) + 3 * threadIdx.x;
    __builtin_prefetch(tr, 0, 0);           // global_prefetch_b8 on gfx1250
    float lx = lows[0],  ly = lows[1],  lm = lows[2];
    float hx = highs[0], hy = highs[1], hm = highs[2];
    float xg = (tr[0] - lx) / (hx - lx);
    float yg = (tr[1] - ly) / (hy - ly);
    float mg = (log10f(tr[2]) - lm) / (hm - lm);
    int xi = (int)floorf(xg * (float)L_);
    int yi = (int)floorf(yg * (float)L_);
    int mi = (int)floorf(mg * (float)NMC_);
    if (mi == mc) {
      int local = yi * L_ + xi - half * HALF;
      if (local >= 0 && local < HALF)       // out-of-tile -> dropped (JAX OOB semantics)
        ((float*)s_half)[local] = pv;       // duplicates write identical values
    }
  }
  __syncthreads();

  v4f* dst = reinterpret_cast<v4f*>(out) + (size_t)plane * (PLANE / 4)
                                         + (size_t)half * HALF_V4;
#if USE_ASYNC_LDS
  auto g = (__attribute__((address_space(1))) v4i*)(dst + threadIdx.x);
  auto l = (__attribute__((address_space(3))) v4i*)(&s_half[threadIdx.x]);
#pragma unroll
  for (int i = 0; i < 8; ++i)
    __builtin_amdgcn_global_store_async_from_lds_b128(g + i * 256, l + i * 256, 0, 0);
  __builtin_amdgcn_s_wait_asynccnt(0);
#else
#pragma unroll
  for (int i = 0; i < 8; ++i)
    __builtin_nontemporal_store(s_half[i * 256 + threadIdx.x],
                                dst + i * 256 + threadIdx.x);
#endif
}

// ---------------------------------------------------------------------------
extern "C" void kernel_launch(void* const* d_in, const int* in_sizes, int n_in,
                              void* d_out, int out_size, void* d_ws, size_t ws_size,
                              hipStream_t stream) {
  (void)in_sizes; (void)n_in; (void)out_size; (void)d_ws; (void)ws_size;
  const float* coord = (const float*)d_in[0];
  const float* lows  = (const float*)d_in[1];
  const float* highs = (const float*)d_in[2];
  float* out = (float*)d_out;

  hipLaunchKernelGGL(fused_kernel, dim3(NBLOCKS), dim3(256), 0, stream,
                     coord, lows, highs, out);
}